// GCN_84499186582208
// MI455X (gfx1250) — compile-verified
//
#include <hip/hip_runtime.h>

typedef __attribute__((ext_vector_type(2))) float v2f;
typedef __attribute__((ext_vector_type(8))) float v8f;

#define IN_F  64
#define HID_F 64
#define OUT_F 32

// ---------------------------------------------------------------------------
// Zero-fill (float4 wide, grid-stride). n must be a multiple of 4 (64 per node).
// ---------------------------------------------------------------------------
__global__ void gcn_zero_kernel(float4* __restrict__ p, long n4) {
    long i = (long)blockIdx.x * blockDim.x + threadIdx.x;
    long stride = (long)gridDim.x * blockDim.x;
    float4 z = make_float4(0.f, 0.f, 0.f, 0.f);
    for (; i < n4; i += stride) p[i] = z;
}

// ---------------------------------------------------------------------------
// Edge scatter-add: one wave (32 lanes) per edge, lane l handles feature
// columns 2l and 2l+1 (float2 gather, two global_atomic_add_f32).
// Gathers and atomics stay resident in the 192MB L2 (node set = 25.6MB).
// ---------------------------------------------------------------------------
template<int F>
__global__ void gcn_scatter_add_kernel(const float* __restrict__ feat,
                                       const int*   __restrict__ src,
                                       const int*   __restrict__ dst,
                                       float*       __restrict__ agg,
                                       int n_edges) {
    int lane = threadIdx.x & 31;
    int wave = blockIdx.x * (blockDim.x >> 5) + (threadIdx.x >> 5);
    int waves_total = gridDim.x * (blockDim.x >> 5);
    int c = lane * 2;
    for (int e = wave; e < n_edges; e += waves_total) {
        int s = __builtin_amdgcn_readfirstlane(src[e]);
        int d = __builtin_amdgcn_readfirstlane(dst[e]);
        const v2f v = *(const v2f*)(feat + (long)s * F + c);
        float* a = agg + (long)d * F + c;
        atomicAdd(a + 0, v.x);
        atomicAdd(a + 1, v.y);
    }
}

// ---------------------------------------------------------------------------
// Dense layer via V_WMMA_F32_16X16X4_F32 (native fp32 matrix pipe).
// One wave owns a 16-row strip x (NTILES*16) cols; K looped in steps of 4.
//
// fp32 WMMA operand layouts (ISA 7.12.2, 32-bit):
//   A 16x4 : lane l -> M = l&15 ; VGPR0 = K=k0+2*(l>>4), VGPR1 = K+1
//   B 4x16 : lane l -> N = l&15 ; same K halving
//   C/D    : VGPR r -> M = r + 8*(l>>4), N = l&15
// ---------------------------------------------------------------------------
template<int K, int NTILES, bool RELU>
__global__ void gcn_gemm_bias_kernel(const float* __restrict__ A,    // n_rows x K
                                     const float* __restrict__ W,    // K x N
                                     const float* __restrict__ bias, // N
                                     float*       __restrict__ C,    // n_rows x N
                                     int n_rows) {
    constexpr int N = NTILES * 16;
    int lane = threadIdx.x & 31;
    int wave = blockIdx.x * (blockDim.x >> 5) + (threadIdx.x >> 5);

    int m0 = wave * 16;               // wave-uniform: no intra-wave divergence
    if (m0 >= n_rows) return;

    int m     = lane & 15;            // row within tile (A) / col within tile (B,C)
    int khalf = (lane >> 4) << 1;     // 0 or 2: which K-pair this half-wave holds

    v8f acc[NTILES];
    v8f zero = {};
#pragma unroll
    for (int t = 0; t < NTILES; ++t) acc[t] = zero;

    const float* arow = A + (long)(m0 + m) * K;

#pragma unroll
    for (int k0 = 0; k0 < K; k0 += 4) {
        // A fragment: contiguous, 8B-aligned float2 per lane
        v2f a = *(const v2f*)(arow + k0 + khalf);
#pragma unroll
        for (int t = 0; t < NTILES; ++t) {
            int n = t * 16 + m;
            v2f b;
            b.x = W[(k0 + khalf) * N + n];
            b.y = W[(k0 + khalf + 1) * N + n];
            acc[t] = __builtin_amdgcn_wmma_f32_16x16x4_f32(
                /*neg_a=*/false, a, /*neg_b=*/false, b,
                /*c_mod=*/(short)0, acc[t],
                /*reuse_a=*/false, /*reuse_b=*/false);
        }
    }

    // Store D + bias (+ReLU). Half-wave writes 16 contiguous floats per row.
    int mrow_off = (lane >> 4) * 8;
#pragma unroll
    for (int t = 0; t < NTILES; ++t) {
        int col = t * 16 + m;
        float bv = bias[col];
#pragma unroll
        for (int r = 0; r < 8; ++r) {
            int row = m0 + r + mrow_off;
            float v = acc[t][r] + bv;
            if (RELU) v = fmaxf(v, 0.f);
            C[(long)row * N + col] = v;
        }
    }
}

// ---------------------------------------------------------------------------
// Orchestration: zero -> scatter(features) -> WMMA+ReLU -> zero ->
//                scatter(h) -> WMMA -> out
// Workspace: agg (n_nodes*64 f32, reused for both layers) + h (n_nodes*64 f32)
// ---------------------------------------------------------------------------
extern "C" void kernel_launch(void* const* d_in, const int* in_sizes, int n_in,
                              void* d_out, int out_size, void* d_ws, size_t ws_size,
                              hipStream_t stream) {
    const float* feat = (const float*)d_in[0];
    const int*   src  = (const int*)d_in[1];
    const int*   dst  = (const int*)d_in[2];
    const float* W1   = (const float*)d_in[3];
    const float* b1   = (const float*)d_in[4];
    const float* W2   = (const float*)d_in[5];
    const float* b2   = (const float*)d_in[6];
    float* out = (float*)d_out;

    int n_nodes = in_sizes[0] / IN_F;
    int n_edges = in_sizes[1];

    float* agg = (float*)d_ws;                        // n_nodes * 64
    float* h   = agg + (size_t)n_nodes * HID_F;       // n_nodes * 64

    long  zn4 = (long)n_nodes * IN_F / 4;
    dim3  blk(256);

    int row_tiles = (n_nodes + 15) / 16;
    int gemm_blocks = (row_tiles + 7) / 8;            // 8 waves per 256-thread block

    // Layer 1
    gcn_zero_kernel<<<1024, blk, 0, stream>>>((float4*)agg, zn4);
    gcn_scatter_add_kernel<IN_F><<<8192, blk, 0, stream>>>(feat, src, dst, agg, n_edges);
    gcn_gemm_bias_kernel<IN_F, HID_F / 16, true>
        <<<gemm_blocks, blk, 0, stream>>>(agg, W1, b1, h, n_nodes);

    // Layer 2 (reuse agg buffer)
    gcn_zero_kernel<<<1024, blk, 0, stream>>>((float4*)agg, zn4);
    gcn_scatter_add_kernel<HID_F><<<8192, blk, 0, stream>>>(h, src, dst, agg, n_edges);
    gcn_gemm_bias_kernel<HID_F, OUT_F / 16, false>
        <<<gemm_blocks, blk, 0, stream>>>(agg, W2, b2, out, n_nodes);
}